// EM_rec_loss_simplify_32555852104028
// MI455X (gfx1250) — compile-verified
//
#include <hip/hip_runtime.h>

typedef float v2f __attribute__((ext_vector_type(2)));
typedef float v4f __attribute__((ext_vector_type(4)));
typedef float v8f __attribute__((ext_vector_type(8)));

namespace {
constexpr int kB = 4, kF = 6, kNS = 7, kNB = 8, kH = 128, kW = 128;
constexpr int kBF = kB * kF;
constexpr int kHW = kH * kW;
constexpr int kChunks  = 32;                 // pixel chunks per (b,f) image
constexpr int kWaves   = 8;
constexpr int kBlock   = kWaves * 32;        // 256 threads = 8 waves (wave32)
constexpr int kPxChunk = kHW / kChunks;      // 512 px per workgroup
constexpr int kPxWave  = kPxChunk / kWaves;  // 64 px per wave
constexpr int kDSteps  = kPxWave / 8;        // 8 double-steps (8 px -> 2 WMMA k-steps)
constexpr float kLn2 = 0.69314718055994530942f;
}

__global__ void em_loss_init(double* ws) { ws[0] = 0.0; }

// Clamped log from raw v_log_f32 (log2) + one mul: 3 VALU total.
// log2(0) = -inf -> *ln2 = -inf -> fmax -> -100, matching the reference clamp.
__device__ __forceinline__ float flog(float x) {
    return fmaxf(__builtin_amdgcn_logf(x) * kLn2, -100.f);
}

__global__ __launch_bounds__(kBlock)
void em_loss_main(const float* __restrict__ seg,    // (bf, 8, HW)
                  const float* __restrict__ masks,  // (bf, 7, HW)
                  const float* __restrict__ rec,    // (bf, 8, 3, HW)
                  const float* __restrict__ tgt,    // (bf, 3, HW)
                  const float* __restrict__ mvis,   // (bf, 7, HW)
                  const float* __restrict__ attn,   // (bf, 7, 8)
                  double* __restrict__ acc_ws)
{
    const int bfi   = blockIdx.x / kChunks;
    const int chunk = blockIdx.x % kChunks;
    const int lane  = threadIdx.x & 31;
    const int wave  = threadIdx.x >> 5;
    const int half  = lane >> 4;      // K-half selector
    const int l16   = lane & 15;      // A: slot row / B: buffer column

    const int px0 = chunk * kPxChunk + wave * kPxWave;

    // Clamp rows so every lane issues valid (broadcast-coalesced) loads.
    // Padded A rows (m>=7) / B columns (n>=8) only write C rows/columns that the
    // final fold never reads, so no zeroing of the operands is needed.
    const int   srow = (l16 < kNS) ? l16 : 0;
    const int   brow = (l16 < kNB) ? l16 : 0;
    const float bval = (l16 < kNB) ? 1.f : 0.f;   // gates s1acc only

    const float* mp  = masks + ((size_t)bfi * kNS + srow) * kHW;
    const float* vp  = mvis  + ((size_t)bfi * kNS + srow) * kHW;
    const float* sp  = seg   + ((size_t)bfi * kNB + brow) * kHW;
    const float* rp0 = rec   + (((size_t)bfi * kNB + brow) * 3 + 0) * kHW;
    const float* rp1 = rec   + (((size_t)bfi * kNB + brow) * 3 + 1) * kHW;
    const float* rp2 = rec   + (((size_t)bfi * kNB + brow) * 3 + 2) * kHW;
    const float* tp0 = tgt   + ((size_t)bfi * 3 + 0) * kHW;
    const float* tp1 = tgt   + ((size_t)bfi * 3 + 1) * kHW;
    const float* tp2 = tgt   + ((size_t)bfi * 3 + 2) * kHW;

    v8f accT = {};      // Σ_px t[s,px] * (log_p - log_1mp)[n,px]
    v8f accM = {};      // Σ_px mvis[s,px] * E[n,px]
    float s1acc = 0.f;  // Σ_px log_1mp for this lane's buffer

    // Pixel->K mapping per double-step d (base = px0 + 8d):
    //   k-step 0: K0,K1 = base+0,1 (half0 .xy)   K2,K3 = base+4,5 (half1 .xy)
    //   k-step 1: K0,K1 = base+2,3 (half0 .zw)   K2,K3 = base+6,7 (half1 .zw)
    // A and B use the identical mapping, so the pixel sum is unchanged.
    for (int d = 0; d < kDSteps; ++d) {
        const int px = px0 + d * 8 + half * 4;   // 16B-aligned float4 per lane

        const v4f m4 = *(const v4f*)(mp  + px);
        const v4f w4 = *(const v4f*)(vp  + px);
        const v4f s4 = *(const v4f*)(sp  + px);
        const v4f r0 = *(const v4f*)(rp0 + px);
        const v4f r1 = *(const v4f*)(rp1 + px);
        const v4f r2 = *(const v4f*)(rp2 + px);
        const v4f t0 = *(const v4f*)(tp0 + px);
        const v4f t1 = *(const v4f*)(tp1 + px);
        const v4f t2 = *(const v4f*)(tp2 + px);

        // ---- A operands: thresholded masks / masks_vis ----
        v2f aT0, aT1, aM0, aM1;
        aT0.x = (m4.x > 0.5f) ? 1.f : 0.f;
        aT0.y = (m4.y > 0.5f) ? 1.f : 0.f;
        aT1.x = (m4.z > 0.5f) ? 1.f : 0.f;
        aT1.y = (m4.w > 0.5f) ? 1.f : 0.f;
        aM0.x = (w4.x > 0.5f) ? 1.f : 0.f;
        aM0.y = (w4.y > 0.5f) ? 1.f : 0.f;
        aM1.x = (w4.z > 0.5f) ? 1.f : 0.f;
        aM1.y = (w4.w > 0.5f) ? 1.f : 0.f;

        // ---- B operand 1: L = clamp(log p) - clamp(log(1-p)) ----
        // 1-s is exact for s>=0.5 (Sterbenz); for s<0.5 abs log error <= 2^-24.
        const float l1x = flog(1.f - s4.x), l1y = flog(1.f - s4.y);
        const float l1z = flog(1.f - s4.z), l1w = flog(1.f - s4.w);
        s1acc += bval * ((l1x + l1y) + (l1z + l1w));
        v2f bT0, bT1;
        bT0.x = flog(s4.x) - l1x;
        bT0.y = flog(s4.y) - l1y;
        bT1.x = flog(s4.z) - l1z;
        bT1.y = flog(s4.w) - l1w;

        // ---- B operand 2: E = sum_c (rec - tgt)^2 ----
        v4f d0, d1, d2;
        d0.x = r0.x - t0.x; d0.y = r0.y - t0.y; d0.z = r0.z - t0.z; d0.w = r0.w - t0.w;
        d1.x = r1.x - t1.x; d1.y = r1.y - t1.y; d1.z = r1.z - t1.z; d1.w = r1.w - t1.w;
        d2.x = r2.x - t2.x; d2.y = r2.y - t2.y; d2.z = r2.z - t2.z; d2.w = r2.w - t2.w;
        v2f bM0, bM1;
        bM0.x = d0.x * d0.x + d1.x * d1.x + d2.x * d2.x;
        bM0.y = d0.y * d0.y + d1.y * d1.y + d2.y * d2.y;
        bM1.x = d0.z * d0.z + d1.z * d1.z + d2.z * d2.z;
        bM1.y = d0.w * d0.w + d1.w * d1.w + d2.w * d2.w;

        // ---- two K=4 steps per double-step, f32 in / f32 acc (exact) ----
        accT = __builtin_amdgcn_wmma_f32_16x16x4_f32(
            false, aT0, false, bT0, (short)0, accT, false, false);
        accM = __builtin_amdgcn_wmma_f32_16x16x4_f32(
            false, aM0, false, bM0, (short)0, accM, false, false);
        accT = __builtin_amdgcn_wmma_f32_16x16x4_f32(
            false, aT1, false, bT1, (short)0, accT, false, false);
        accM = __builtin_amdgcn_wmma_f32_16x16x4_f32(
            false, aM1, false, bM1, (short)0, accM, false, false);
    }

    // ---- fold attn_index: lane n (0..7) holds column buffer n, vgpr m = slot m ----
    const float invHW = 1.0f / (float)kHW;
    float contrib = 0.f;
    if (l16 < kNB) {
        const float* ap = attn + (size_t)bfi * kNS * kNB + l16;
        float sum = 0.f, aib = 0.f;
        #pragma unroll
        for (int m = 0; m < kNS; ++m) {
            float a = ap[m * kNB];
            aib += a;
            sum += a * (0.1f * accM[m] - accT[m] * invHW);
        }
        // half==1 lanes hold rows m=8..15 of C (garbage/unused); only their S1 term counts.
        contrib = (half == 0 ? sum : 0.f) + (-invHW) * aib * s1acc;
    }

    // ---- wave reduction, one f64 atomic per wave ----
    #pragma unroll
    for (int off = 16; off > 0; off >>= 1)
        contrib += __shfl_xor(contrib, off, 32);
    if (lane == 0)
        unsafeAtomicAdd(acc_ws, (double)contrib);
}

__global__ void em_loss_finalize(const double* __restrict__ ws,
                                 float* __restrict__ out)
{
    // total / (b*f*n_slots*n_buffer) * LOSS_WEIGHT
    out[0] = (float)(ws[0] * (20.0 / (4.0 * 6.0 * 7.0 * 8.0)));
}

extern "C" void kernel_launch(void* const* d_in, const int* in_sizes, int n_in,
                              void* d_out, int out_size, void* d_ws, size_t ws_size,
                              hipStream_t stream) {
    const float* seg   = (const float*)d_in[0];  // segmentations   (4,6,8,128,128)
    const float* masks = (const float*)d_in[1];  // masks           (4,6,7,128,128)
    const float* rec   = (const float*)d_in[2];  // reconstructions (4,6,8,3,128,128)
    const float* tgt   = (const float*)d_in[3];  // rec_tgt         (4,6,3,128,128)
    const float* mvis  = (const float*)d_in[4];  // masks_vis       (4,6,7,128,128)
    const float* attn  = (const float*)d_in[5];  // attn_index      (4,6,7,8)
    // d_in[6], d_in[7] (slots, pred_slots) unused by the reference.

    double* acc = (double*)d_ws;
    float*  out = (float*)d_out;

    em_loss_init<<<1, 1, 0, stream>>>(acc);
    em_loss_main<<<kBF * kChunks, kBlock, 0, stream>>>(
        seg, masks, rec, tgt, mvis, attn, acc);
    em_loss_finalize<<<1, 1, 0, stream>>>(acc, out);
}